// GroupBRouter_78288663872329
// MI455X (gfx1250) — compile-verified
//
#include <hip/hip_runtime.h>

typedef __attribute__((ext_vector_type(2))) float v2f;
typedef __attribute__((ext_vector_type(4))) float v4f;
typedef __attribute__((ext_vector_type(8))) float v8f;

#define B_DIM 16
#define N_DIM 4096
#define D_DIM 1024
#define E_DIM 64
#define TOPK 2
#define KCHUNK 64                // 16 WMMA k-steps of 4
#define TOK_PER_WAVE 16
#define WAVES_PER_BLOCK 8
#define TOK_PER_BLOCK (TOK_PER_WAVE * WAVES_PER_BLOCK)  // 128
#define LG_PITCH 68              // padded row (floats): conflict-free + 16B aligned

__global__ __launch_bounds__(256) void
router_wmma_kernel(const float* __restrict__ tokens,
                   const int*   __restrict__ tvec,
                   const float* __restrict__ Wg,
                   const float* __restrict__ bg,
                   float*       __restrict__ out)
{
    // W chunk in B-fragment layout: [16 ksteps][4 expert tiles][32 lanes] of float2
    __shared__ v2f   Wbuf[16 * 4 * 32];                       // 16 KB
    __shared__ float lg[WAVES_PER_BLOCK][TOK_PER_WAVE][LG_PITCH]; // ~34 KB

    const int tid  = threadIdx.x;
    const int wave = tid >> 5;
    const int lane = tid & 31;
    const int n15  = lane & 15;
    const int koff = (lane >> 4) << 1;      // 0 or 2

    const int tok0  = blockIdx.x * TOK_PER_BLOCK + wave * TOK_PER_WAVE;
    const int batch = tok0 / N_DIM;         // block never straddles a batch

    // ---- accumulators: 16x16 f32 C tiles, pre-loaded with bias b_g[e] ----
    const float b0 = bg[ 0 + n15];
    const float b1 = bg[16 + n15];
    const float b2 = bg[32 + n15];
    const float b3 = bg[48 + n15];
    v8f acc0, acc1, acc2, acc3;
#pragma unroll
    for (int r = 0; r < 8; ++r) { acc0[r] = b0; acc1[r] = b1; acc2[r] = b2; acc3[r] = b3; }

    // A fragment source: lane l reads token row (l&15), columns k0 + 2*(l>=16) + {0,1}
    const float* Arow = tokens + (size_t)(tok0 + n15) * D_DIM + koff;

    for (int kc = 0; kc < D_DIM; kc += KCHUNK) {
        __syncthreads();
        // Cooperative stage of W chunk into LDS in fragment order.
        // entry i: s = i>>7 (kstep), t = (i>>5)&3 (expert tile), l = i&31 (lane)
#pragma unroll
        for (int i = tid; i < 16 * 4 * 32; i += 256) {
            const int s = i >> 7;
            const int t = (i >> 5) & 3;
            const int l = i & 31;
            const int e = t * 16 + (l & 15);
            const int k = kc + s * 4 + ((l >> 4) << 1);
            Wbuf[i] = *(const v2f*)(Wg + (size_t)e * D_DIM + k);
        }
        __syncthreads();

#pragma unroll
        for (int s = 0; s < 16; ++s) {
            const v2f a = *(const v2f*)(Arow + kc + s * 4);
            const v2f w0 = Wbuf[(s * 4 + 0) * 32 + lane];
            const v2f w1 = Wbuf[(s * 4 + 1) * 32 + lane];
            const v2f w2 = Wbuf[(s * 4 + 2) * 32 + lane];
            const v2f w3 = Wbuf[(s * 4 + 3) * 32 + lane];
            acc0 = __builtin_amdgcn_wmma_f32_16x16x4_f32(false, a, false, w0, (short)0, acc0, false, false);
            acc1 = __builtin_amdgcn_wmma_f32_16x16x4_f32(false, a, false, w1, (short)0, acc1, false, false);
            acc2 = __builtin_amdgcn_wmma_f32_16x16x4_f32(false, a, false, w2, (short)0, acc2, false, false);
            acc3 = __builtin_amdgcn_wmma_f32_16x16x4_f32(false, a, false, w3, (short)0, acc3, false, false);
        }
    }

    // ---- spill logits to LDS: lane l, reg r -> token m = r + 8*(l>=16), expert t*16 + (l&15)
    const int mhalf = (lane >> 4) << 3;
#pragma unroll
    for (int r = 0; r < 8; ++r) {
        lg[wave][mhalf + r][ 0 + n15] = acc0[r];
        lg[wave][mhalf + r][16 + n15] = acc1[r];
        lg[wave][mhalf + r][32 + n15] = acc2[r];
        lg[wave][mhalf + r][48 + n15] = acc3[r];
    }
    __syncthreads();

    // ---- per-token epilogue: softmax -> floor -> hard cap -> top-2 gates ----
    const float cap = 0.5f + 1.1f * ((float)tvec[batch] / 1000.0f);  // CAP_LOW + (CAP_HIGH+CAP_LOW)*t/Tmax
    if (lane < 16) {
        float* row = lg[wave][lane];

        float mx = row[0];
#pragma unroll 8
        for (int e = 1; e < E_DIM; ++e) mx = fmaxf(mx, row[e]);

        float sum = 0.f;
#pragma unroll 8
        for (int e = 0; e < E_DIM; ++e) { float ex = __expf(row[e] - mx); row[e] = ex; sum += ex; }
        const float inv = 1.0f / sum;

        // floor mix: alpha = 0.15 -> p = 0.85*p + 0.15/64 ; then cap sums
        float excess_sum = 0.f, head_sum = 0.f;
#pragma unroll 8
        for (int e = 0; e < E_DIM; ++e) {
            float p = 0.85f * (row[e] * inv) + (0.15f / 64.0f);
            float capped = fminf(p, cap);
            excess_sum += p - capped;
            head_sum   += cap - capped;
            row[e] = capped;
        }
        const float factor = excess_sum / fmaxf(head_sum, 1e-8f);

        float v1 = -1.0f, v2 = -1.0f; int i1 = 0, i2 = 0;
#pragma unroll 8
        for (int e = 0; e < E_DIM; ++e) {
            float capped = row[e];
            float fin = capped + (cap - capped) * factor;
            row[e] = fin;
            if (fin > v1)      { v2 = v1; i2 = i1; v1 = fin; i1 = e; }
            else if (fin > v2) { v2 = fin; i2 = e; }
        }
        // sparse top-2 gate row
#pragma unroll 8
        for (int e = 0; e < E_DIM; ++e)
            row[e] = (e == i1) ? v1 : ((e == i2) ? v2 : 0.0f);
    }
    __syncthreads();

    // ---- coalesced stores: 2 tokens per pass, float4 per lane ----
#pragma unroll
    for (int p = 0; p < 8; ++p) {
        const int m  = p * 2 + (lane >> 4);
        const int e0 = n15 * 4;
        const v4f g  = *(const v4f*)&lg[wave][m][e0];
        *(v4f*)(out + (size_t)(tok0 + m) * E_DIM + e0) = g;
    }
}

extern "C" void kernel_launch(void* const* d_in, const int* in_sizes, int n_in,
                              void* d_out, int out_size, void* d_ws, size_t ws_size,
                              hipStream_t stream) {
    const float* tokens = (const float*)d_in[0];   // (B,N,D) f32
    const int*   tvec   = (const int*)d_in[1];     // (B,)    i32
    const float* Wg     = (const float*)d_in[2];   // (E,D)   f32
    const float* bg     = (const float*)d_in[3];   // (E,)    f32
    float*       out    = (float*)d_out;           // (B,N,E) f32

    const int total_tokens = B_DIM * N_DIM;        // 65536
    dim3 grid(total_tokens / TOK_PER_BLOCK);       // 512
    dim3 block(256);
    router_wmma_kernel<<<grid, block, 0, stream>>>(tokens, tvec, Wg, bg, out);
}